// SSIMMetric_59571196395809
// MI455X (gfx1250) — compile-verified
//
#include <hip/hip_runtime.h>

typedef __attribute__((ext_vector_type(2))) float v2f;
typedef __attribute__((ext_vector_type(8))) float v8f;

#define IMG_W 512
#define IMG_H 512
#define TILE  16
#define RAD   5
#define REG   26           // TILE + 2*RAD
#define KPAD  28           // REG padded to 7 K-chunks of 4 (cols 26,27 are zero)
#define SSIM_C1 1.0e-4f
#define SSIM_C2 9.0e-4f
#define NELEM  (REG * REG) // 676
#define NITER  22          // ceil(676/32)

// 11-tap Gaussian, sigma = 1.5, normalized (precomputed). Used once per block.
__device__ __forceinline__ float gtap(int d) {
    const float G[11] = {0.00102828f, 0.00759770f, 0.03599700f, 0.10934930f,
                         0.21298400f, 0.26598350f, 0.21298400f, 0.10934930f,
                         0.03599700f, 0.00759770f, 0.00102828f};
    float r = 0.0f;
#pragma unroll
    for (int t = 0; t < 11; ++t) r = (d == t) ? G[t] : r;
    return r;
}

__global__ __launch_bounds__(32)
void ssim_wmma_kernel(const float* __restrict__ pred,
                      const float* __restrict__ targ,
                      float* __restrict__ out) {
    // Stat plane S: 26 rows x 28 cols (cols 26,27 zero) -> unconditional K=28 GEMM.
    __shared__ __align__(16) float Sl[REG * KPAD];
    // Horizontal result, TRANSPOSED: Hlt[col][row], 16 x 28 (rows 26,27 zero).
    __shared__ __align__(16) float Hlt[16 * KPAD];
    // Zero-padded Gaussian band table: Gt[16+d] = g[d], d in [0,10].
    __shared__ __align__(16) float Gt[64];

    const int lane  = threadIdx.x;      // one wave per block
    const int tile  = blockIdx.x;       // 48 planes * 32*32 tiles
    const int plane = tile >> 10;
    const int rem   = tile & 1023;
    const int ty    = rem >> 5;
    const int tx    = rem & 31;
    const int base  = plane * (IMG_W * IMG_H);
    const int iy0   = ty * TILE - RAD;
    const int ix0   = tx * TILE - RAD;

    // ---- one-time LDS init: weight table + pad columns ----
    Gt[lane] = 0.f;
    Gt[lane + 32] = 0.f;
    for (int r = lane; r < REG; r += 32) {           // Sl pad cols stay zero forever
        Sl[r * KPAD + REG]     = 0.f;
        Sl[r * KPAD + REG + 1] = 0.f;
    }
    if (lane < 16) {                                  // Hlt pad cols stay zero forever
        Hlt[lane * KPAD + REG]     = 0.f;
        Hlt[lane * KPAD + REG + 1] = 0.f;
    }
    __syncthreads();
    if (lane < 11) Gt[16 + lane] = gtap(lane);
    __syncthreads();

    // ---- stage 1: pred/target halo region into registers (branch-free) ----
    float pv[NITER], tv[NITER];
#pragma unroll
    for (int ii = 0; ii < NITER; ++ii) {
        int i = lane + ii * 32;
        int r = i / REG, c = i - r * REG;
        int y = iy0 + r, x = ix0 + c;
        bool inb = ((unsigned)y < IMG_H) && ((unsigned)x < IMG_W) && (i < NELEM);
        float ok = inb ? 1.f : 0.f;
        int yc = min(max(y, 0), IMG_H - 1);
        int xc = min(max(x, 0), IMG_W - 1);
        int off = base + yc * IMG_W + xc;
        pv[ii] = pred[off] * ok;
        tv[ii] = targ[off] * ok;
    }

    // ---- constant banded-Gaussian WMMA fragments from the LDS table ----
    // wch[kc] serves as B of the horizontal pass (Wh[k][j]=g[k-j]) and as
    // A of the vertical pass (Wv[i][k]=g[k-i]) -- identical per-lane values
    // under the symmetric 16x4 / 4x16 f32 register layouts.
    const int sub = (lane >> 4) << 1;   // K sub-offset per half-wave (0 or 2)
    const int mn  = lane & 15;          // M (A-frag) / N (B-frag) index
    v2f wch[7];
#pragma unroll
    for (int kc = 0; kc < 7; ++kc) {
        int q = 16 + kc * 4 + sub - mn;   // in [1, 43]
        wch[kc].x = Gt[q];
        wch[kc].y = Gt[q + 1];
    }

    v8f res[5];

#pragma unroll
    for (int s = 0; s < 5; ++s) {
        // ---- build statistic plane in LDS from registers ----
#pragma unroll
        for (int ii = 0; ii < NITER; ++ii) {
            int i = lane + ii * 32;
            if (ii < NITER - 1 || i < NELEM) {   // guard folds away except last iter
                int r = i / REG, c = i - r * REG;
                float p = pv[ii], t = tv[ii];
                float v;
                if      (s == 0) v = p;
                else if (s == 1) v = t;
                else if (s == 2) v = p * p;
                else if (s == 3) v = t * t;
                else             v = p * t;
                Sl[r * KPAD + c] = v;
            }
        }
        __syncthreads();

        // ---- horizontal pass: H(26x16) = S(26x28) x Wh(28x16), two row groups ----
#pragma unroll
        for (int rg = 0; rg < 2; ++rg) {
            const int rowbase = rg * 10;            // rows 0..15 and 10..25
            v8f acc = {0.f,0.f,0.f,0.f,0.f,0.f,0.f,0.f};
#pragma unroll
            for (int kc = 0; kc < 7; ++kc) {
                int kb = kc * 4 + sub;              // even -> 8B aligned pair
                v2f a = *(const v2f*)&Sl[(rowbase + mn) * KPAD + kb];
                acc = __builtin_amdgcn_wmma_f32_16x16x4_f32(
                          false, a, false, wch[kc], (short)0, acc, false, false);
            }
            const int hrow = rowbase + ((lane >> 4) << 3);  // D: M = v + 8*(lane/16)
#pragma unroll
            for (int v = 0; v < 8; ++v)             // contiguous -> b64/b128 merges
                Hlt[mn * KPAD + hrow + v] = acc[v];
        }
        __syncthreads();

        // ---- vertical pass: D(16x16) = Wv(16x28) x H(28x16) ----
        v8f acc = {0.f,0.f,0.f,0.f,0.f,0.f,0.f,0.f};
#pragma unroll
        for (int kc = 0; kc < 7; ++kc) {
            int kb = kc * 4 + sub;
            v2f b = *(const v2f*)&Hlt[mn * KPAD + kb];   // transposed H: contiguous
            acc = __builtin_amdgcn_wmma_f32_16x16x4_f32(
                      false, wch[kc], false, b, (short)0, acc, false, false);
        }
        res[s] = acc;
        __syncthreads();   // before next stat overwrites Sl/Hlt
    }

    // ---- SSIM map (all 5 results share the D-matrix layout) + partial mean ----
    float sum = 0.f;
#pragma unroll
    for (int v = 0; v < 8; ++v) {
        float mu1 = res[0][v], mu2 = res[1][v];
        float ex2 = res[2][v], ey2 = res[3][v], exy = res[4][v];
        float mu1sq = mu1 * mu1, mu2sq = mu2 * mu2, mu12 = mu1 * mu2;
        float s1 = ex2 - mu1sq, s2 = ey2 - mu2sq, s12 = exy - mu12;
        float num = (2.f * mu12 + SSIM_C1) * (2.f * s12 + SSIM_C2);
        float den = (mu1sq + mu2sq + SSIM_C1) * (s1 + s2 + SSIM_C2);
        sum += num / den;
    }
#pragma unroll
    for (int off = 16; off >= 1; off >>= 1)
        sum += __shfl_xor(sum, off, 32);

    if (lane == 0) {
        const float invN = 1.0f / (16.0f * 3.0f * 512.0f * 512.0f);
        atomicAdd(out, sum * invN);
    }
}

__global__ void ssim_zero_kernel(float* out) {
    if (threadIdx.x == 0) out[0] = 0.f;
}

extern "C" void kernel_launch(void* const* d_in, const int* in_sizes, int n_in,
                              void* d_out, int out_size, void* d_ws, size_t ws_size,
                              hipStream_t stream) {
    const float* pred = (const float*)d_in[0];
    const float* targ = (const float*)d_in[1];
    float* out = (float*)d_out;

    ssim_zero_kernel<<<1, 32, 0, stream>>>(out);

    const int nblocks = 48 * 32 * 32;   // (B*C) planes * (512/16)^2 tiles
    ssim_wmma_kernel<<<nblocks, 32, 0, stream>>>(pred, targ, out);
}